// BahdanauAttention_12970801233975
// MI455X (gfx1250) — compile-verified
//
#include <hip/hip_runtime.h>
#include <hip/hip_bf16.h>

typedef __attribute__((ext_vector_type(2))) float v2f;
typedef __attribute__((ext_vector_type(8))) float v8f;
typedef int v4i __attribute__((vector_size(16)));

#if __has_builtin(__builtin_amdgcn_global_load_async_to_lds_b128)
#define USE_ASYNC_LDS 1
#else
#define USE_ASYNC_LDS 0
#endif

__device__ inline __attribute__((address_space(1))) v4i* as_global(const void* p) {
    return (__attribute__((address_space(1))) v4i*)(size_t)p;
}
__device__ inline __attribute__((address_space(3))) v4i* as_lds(const void* p) {
    // flat pointers to LDS carry the LDS offset in the low 32 bits
    return (__attribute__((address_space(3))) v4i*)(unsigned)(size_t)p;
}
__device__ inline void wait_async_lds() {
#if USE_ASYNC_LDS
#if __has_builtin(__builtin_amdgcn_s_wait_asynccnt)
    __builtin_amdgcn_s_wait_asynccnt(0);
#else
    asm volatile("s_wait_asynccnt 0x0" ::: "memory");
#endif
#endif
}

// ---------------------------------------------------------------------------
// Batched GEMM, row-major: C = A(M,K) @ B(K,N); lda=K, ldb=ldc=N.
// M,N multiples of 64; K multiple of 32.
// 128 threads = 4 waves (2x2); macro tile 64x64; each wave owns a 32x32 tile
// (4 accumulators) via V_WMMA_F32_16X16X4_F32. Tiles double-buffered in LDS,
// filled with GLOBAL_LOAD_ASYNC_TO_LDS_B128 when available.
// act = 1 -> apply tanh on store.
// ---------------------------------------------------------------------------
#define MT 64
#define NT 64
#define KT 32

__global__ __launch_bounds__(128)
void gemm_wmma_f32(const float* __restrict__ Ag, const float* __restrict__ Bg,
                   float* __restrict__ Cg, int M, int N, int K,
                   long long sA, long long sB, long long sC, int act)
{
    __shared__ __align__(16) float lA[2][MT][KT + 4];
    __shared__ __align__(16) float lB[2][KT][NT + 4];

    (void)M;
    const int bz = blockIdx.z;
    const float* A = Ag + (long long)bz * sA;
    const float* B = Bg + (long long)bz * sB;
    float*       C = Cg + (long long)bz * sC;

    const int mBase = blockIdx.y * MT;
    const int nBase = blockIdx.x * NT;
    const int tid   = threadIdx.x;
    const int wave  = tid >> 5;
    const int lane  = tid & 31;
    const int wm    = (wave >> 1) * 32;   // wave row offset in macro tile
    const int wn    = (wave & 1) * 32;    // wave col offset in macro tile
    const int col   = lane & 15;
    const int half  = lane >> 4;          // 0 | 1
    const int khalf = half * 2;           // K split across lane halves

    v8f acc00 = {}, acc01 = {}, acc10 = {}, acc11 = {};

    // Stage one 64x32 A tile + 32x64 B tile (8 KB each) into LDS buffer `buf`.
    auto load_tiles = [&](int buf, int k0) {
#pragma unroll
        for (int r = 0; r < 4; ++r) {
            const int c    = tid + r * 128;
            const int arow = c >> 3;             // 64 rows, 8 x 16B chunks/row
            const int ac4  = (c & 7) << 2;
            const int brow = c >> 4;             // 32 rows, 16 x 16B chunks/row
            const int bc4  = (c & 15) << 2;
            const float* ga = &A[(size_t)(mBase + arow) * K + k0 + ac4];
            const float* gb = &B[(size_t)(k0 + brow) * N + nBase + bc4];
#if USE_ASYNC_LDS
            __builtin_amdgcn_global_load_async_to_lds_b128(
                as_global(ga), as_lds(&lA[buf][arow][ac4]), 0, 0);
            __builtin_amdgcn_global_load_async_to_lds_b128(
                as_global(gb), as_lds(&lB[buf][brow][bc4]), 0, 0);
#else
            *(float4*)&lA[buf][arow][ac4] = *(const float4*)ga;
            *(float4*)&lB[buf][brow][bc4] = *(const float4*)gb;
#endif
        }
    };

    load_tiles(0, 0);
#if !USE_ASYNC_LDS
    __syncthreads();
#endif

    int buf = 0;
    for (int k0 = 0; k0 < K; k0 += KT, buf ^= 1) {
#if USE_ASYNC_LDS
        wait_async_lds();       // own async writes for current buffer complete
        __syncthreads();        // all waves' writes visible; prior reads done
        if (k0 + KT < K) load_tiles(buf ^ 1, k0 + KT);   // overlap with WMMA
#endif

#pragma unroll
        for (int kk = 0; kk < KT; kk += 4) {
            v2f a0, a1, b0, b1;
            a0[0] = lA[buf][wm + col][kk + khalf];
            a0[1] = lA[buf][wm + col][kk + khalf + 1];
            a1[0] = lA[buf][wm + 16 + col][kk + khalf];
            a1[1] = lA[buf][wm + 16 + col][kk + khalf + 1];
            b0[0] = lB[buf][kk + khalf][wn + col];
            b0[1] = lB[buf][kk + khalf + 1][wn + col];
            b1[0] = lB[buf][kk + khalf][wn + 16 + col];
            b1[1] = lB[buf][kk + khalf + 1][wn + 16 + col];
            acc00 = __builtin_amdgcn_wmma_f32_16x16x4_f32(false, a0, false, b0, (short)0, acc00, false, false);
            acc01 = __builtin_amdgcn_wmma_f32_16x16x4_f32(false, a0, false, b1, (short)0, acc01, false, false);
            acc10 = __builtin_amdgcn_wmma_f32_16x16x4_f32(false, a1, false, b0, (short)0, acc10, false, false);
            acc11 = __builtin_amdgcn_wmma_f32_16x16x4_f32(false, a1, false, b1, (short)0, acc11, false, false);
        }

#if !USE_ASYNC_LDS
        __syncthreads();
        if (k0 + KT < K) {
            load_tiles(buf ^ 1, k0 + KT);
            __syncthreads();
        }
#endif
    }

    // D layout: VGPR r -> row (base + r) for lanes 0-15, row (base + r + 8) for 16-31
    const int ccol = nBase + wn + col;
#pragma unroll
    for (int r = 0; r < 8; ++r) {
        const int row0 = mBase + wm + r + half * 8;
        float v0 = acc00[r], v1 = acc01[r], v2 = acc10[r], v3 = acc11[r];
        if (act) { v0 = tanhf(v0); v1 = tanhf(v1); v2 = tanhf(v2); v3 = tanhf(v3); }
        C[(size_t)row0 * N + ccol]             = v0;
        C[(size_t)row0 * N + ccol + 16]        = v1;
        C[(size_t)(row0 + 16) * N + ccol]      = v2;
        C[(size_t)(row0 + 16) * N + ccol + 16] = v3;
    }
}

// ---------------------------------------------------------------------------
// Fused score + mask + softmax: one block per (b,t).
// score[s] = sum_h v[h] * tanh(pq[b,t,h] + pe[b,s,h]); masked softmax over s.
// ---------------------------------------------------------------------------
__global__ __launch_bounds__(256)
void score_softmax_kernel(const float* __restrict__ pq, const float* __restrict__ pe,
                          const float* __restrict__ v, const int* __restrict__ lens,
                          float* __restrict__ align_out, int T, int S, int H)
{
    __shared__ float spq[512];
    __shared__ float sv[512];
    __shared__ float ssc[512];
    __shared__ float red[256];

    const int bt  = blockIdx.x;
    const int b   = bt / T;
    const int tid = threadIdx.x;

    for (int i = tid; i < H; i += 256) {
        spq[i] = pq[(size_t)bt * H + i];
        sv[i]  = v[i];
    }
    __syncthreads();

    const int wave = tid >> 5;
    const int lane = tid & 31;
    const float* peB = pe + (size_t)b * S * H;

    for (int s = wave; s < S; s += 8) {
        const float* row = peB + (size_t)s * H;
        if (s + 8 < S)  // prefetch next row (global_prefetch_b8)
            __builtin_prefetch(peB + (size_t)(s + 8) * H + lane * 16, 0, 3);
        float a = 0.0f;
        for (int h = lane; h < H; h += 32)
            a += sv[h] * tanhf(spq[h] + row[h]);
#pragma unroll
        for (int off = 16; off > 0; off >>= 1)
            a += __shfl_xor(a, off, 32);
        if (lane == 0) ssc[s] = a;
    }
    __syncthreads();

    const int len = lens[b];
    const float NEG_INF = -__builtin_inff();
    float x0 = (tid < len) ? ssc[tid] : NEG_INF;
    float x1 = (tid + 256 < len) ? ssc[tid + 256] : NEG_INF;

    red[tid] = fmaxf(x0, x1);
    __syncthreads();
    for (int st = 128; st > 0; st >>= 1) {
        if (tid < st) red[tid] = fmaxf(red[tid], red[tid + st]);
        __syncthreads();
    }
    const float mx = red[0];
    __syncthreads();

    float e0 = (tid < len) ? __expf(x0 - mx) : 0.0f;
    float e1 = (tid + 256 < len) ? __expf(x1 - mx) : 0.0f;
    red[tid] = e0 + e1;
    __syncthreads();
    for (int st = 128; st > 0; st >>= 1) {
        if (tid < st) red[tid] += red[tid + st];
        __syncthreads();
    }
    const float inv = 1.0f / red[0];

    float* arow = align_out + (size_t)bt * S;
    arow[tid]       = e0 * inv;
    arow[tid + 256] = e1 * inv;
}

// ---------------------------------------------------------------------------
// cat[b,t,0:H] = context, cat[b,t,H:2H] = query
// ---------------------------------------------------------------------------
__global__ void concat_kernel(const float* __restrict__ ctx, const float* __restrict__ q,
                              float* __restrict__ cat, int H, long long total)
{
    long long i = (long long)blockIdx.x * blockDim.x + threadIdx.x;
    if (i >= total) return;
    long long bt = i / (2 * H);
    int h = (int)(i % (2 * H));
    cat[i] = (h < H) ? ctx[bt * H + h] : q[bt * H + (h - H)];
}

// ---------------------------------------------------------------------------
extern "C" void kernel_launch(void* const* d_in, const int* in_sizes, int n_in,
                              void* d_out, int out_size, void* d_ws, size_t ws_size,
                              hipStream_t stream)
{
    (void)in_sizes; (void)n_in; (void)out_size; (void)ws_size;

    const float* query = (const float*)d_in[0];   // (B,T,H)
    const float* enc   = (const float*)d_in[1];   // (B,S,H)
    const int*   lens  = (const int*)  d_in[2];   // (B,)
    const float* W_h   = (const float*)d_in[3];   // (H,H)
    const float* W_s   = (const float*)d_in[4];   // (H,H)
    const float* v     = (const float*)d_in[5];   // (H,)
    const float* W_out = (const float*)d_in[6];   // (2H,H)
    float* out = (float*)d_out;                   // (B,T,H)

    const int Bn = 4, T = 128, S = 512, H = 512;

    float* ws    = (float*)d_ws;
    float* pq    = ws;                              // B*T*H
    float* pe    = pq    + (size_t)Bn * T * H;      // B*S*H
    float* align = pe    + (size_t)Bn * S * H;      // B*T*S
    float* ctx   = align + (size_t)Bn * T * S;      // B*T*H
    float* cat   = ctx   + (size_t)Bn * T * H;      // B*T*2H

    const dim3 gblk(128);

    // pq = (B*T,H) @ W_s : M=512 N=512 K=512
    gemm_wmma_f32<<<dim3(512 / NT, 512 / MT, 1), gblk, 0, stream>>>(
        query, W_s, pq, 512, 512, 512, 0, 0, 0, 0);

    // pe = (B*S,H) @ W_h : M=2048 N=512 K=512
    gemm_wmma_f32<<<dim3(512 / NT, 2048 / MT, 1), gblk, 0, stream>>>(
        enc, W_h, pe, 2048, 512, 512, 0, 0, 0, 0);

    // scores + masked softmax -> align (B,T,S)
    score_softmax_kernel<<<dim3(Bn * T), dim3(256), 0, stream>>>(
        pq, pe, v, lens, align, T, S, H);

    // ctx[b] = align[b] (T,S) @ enc[b] (S,H), batched over z
    gemm_wmma_f32<<<dim3(512 / NT, 128 / MT, Bn), gblk, 0, stream>>>(
        align, enc, ctx, 128, 512, 512,
        (long long)T * S, (long long)S * H, (long long)T * H, 0);

    // cat = [ctx, query]
    const long long totalCat = (long long)Bn * T * 2 * H;
    concat_kernel<<<dim3((unsigned)((totalCat + 255) / 256)), dim3(256), 0, stream>>>(
        ctx, query, cat, H, totalCat);

    // out = tanh( cat (512,1024) @ W_out (1024,512) )  -- fused activation
    gemm_wmma_f32<<<dim3(512 / NT, 512 / MT, 1), gblk, 0, stream>>>(
        cat, W_out, out, 512, 512, 1024, 0, 0, 0, 1);
}